// ToHyperSphere_11081015624232
// MI455X (gfx1250) — compile-verified
//
#include <hip/hip_runtime.h>
#include <hip/hip_bf16.h>

#ifndef __has_builtin
#define __has_builtin(x) 0
#endif

// ---------------------------------------------------------------------------
// gfx1250 async global<->LDS data-mover path (ASYNCcnt-tracked).
// Builtin params are AS(1)/AS(3) pointers to 'int __attribute__((vector_size(16)))'.
// ---------------------------------------------------------------------------
#if defined(__AMDGCN__) && \
    __has_builtin(__builtin_amdgcn_global_load_async_to_lds_b128) && \
    __has_builtin(__builtin_amdgcn_global_store_async_from_lds_b128)
#define USE_ASYNC_LDS 1
typedef int v4i __attribute__((__vector_size__(4 * sizeof(int))));
typedef __attribute__((address_space(1))) v4i* as1_v4i_p;
typedef __attribute__((address_space(3))) v4i* as3_v4i_p;
#else
#define USE_ASYNC_LDS 0
#endif

__device__ __forceinline__ void wait_async0() {
#if defined(__AMDGCN__)
#if __has_builtin(__builtin_amdgcn_s_wait_asynccnt)
  __builtin_amdgcn_s_wait_asynccnt(0);
#else
  asm volatile("s_wait_asynccnt 0" ::: "memory");
#endif
#endif
}

__device__ __forceinline__ void wait_ds0() {
#if defined(__AMDGCN__)
#if __has_builtin(__builtin_amdgcn_s_wait_dscnt)
  __builtin_amdgcn_s_wait_dscnt(0);
#else
  asm volatile("s_wait_dscnt 0" ::: "memory");
#endif
#endif
}

// Raw-rate hardware sqrt / rsqrt (v_sqrt_f32 / v_rsq_f32), guarded.
__device__ __forceinline__ float dev_sqrtf(float v) {
#if defined(__AMDGCN__) && __has_builtin(__builtin_amdgcn_sqrtf)
  return __builtin_amdgcn_sqrtf(v);
#else
  return __builtin_sqrtf(v);
#endif
}
__device__ __forceinline__ float dev_rsqf(float v) {
#if defined(__AMDGCN__) && __has_builtin(__builtin_amdgcn_rsqf)
  return __builtin_amdgcn_rsqf(v);
#else
  return 1.0f / __builtin_sqrtf(v);
#endif
}

static constexpr float kPi    = 3.14159265358979323846f;
static constexpr float kTwoPi = 6.28318530717958647692f;

// acos via A&S 4.4.45 (7-term, |err| <= ~2e-8): acos(u) = sqrt(1-u)*poly(u),
// u in [0,1]; negative side folded with pi - acos(-t). ~11 VALU ops.
__device__ __forceinline__ float fast_acosf(float t) {
  const float u = __builtin_fabsf(t);
  float p = fmaf(u, -0.0012624911f, 0.0066700901f);
  p = fmaf(p, u, -0.0170881256f);
  p = fmaf(p, u, 0.0308918810f);
  p = fmaf(p, u, -0.0501743046f);
  p = fmaf(p, u, 0.0889789874f);
  p = fmaf(p, u, -0.2145988016f);
  p = fmaf(p, u, 1.5707963050f);
  const float a = dev_sqrtf(1.0f - u) * p;   // NaN for u>1, matching arccos
  return (t < 0.0f) ? (kPi - a) : a;
}

static constexpr int kD      = 1024;  // row length
static constexpr int kChunk  = 32;    // contiguous elements owned per lane
static constexpr int kWaves  = 8;     // waves (rows) per block
static constexpr int kStride = 36;    // padded words per chunk in LDS

__global__ __launch_bounds__(256) void tohypersphere_kernel(
    const float* __restrict__ x, float* __restrict__ out, int rows) {
  const int lane = threadIdx.x & 31;
  const int wave = threadIdx.x >> 5;
  const int row  = blockIdx.x * kWaves + wave;
  if (row >= rows) return;

  const float* grow = x   + (size_t)row * kD;
  float*       orow = out + (size_t)row * kD;

  float xv[kChunk];

#if USE_ASYNC_LDS
  __shared__ __align__(16) float lds[kWaves * kStride * 32];
  float* ldsBase = &lds[wave * kStride * 32];

  // Coalesced async global->LDS: 8 instructions x (32 lanes x 16B) = 4KB row;
  // 16B granule of element e = s*128 + lane*4 lands in the padded chunk layout.
  #pragma unroll
  for (int s = 0; s < 8; ++s) {
    const int e = s * 128 + lane * 4;
    const int c = e >> 5;   // owner chunk
    const int w = e & 31;   // word within chunk
    __builtin_amdgcn_global_load_async_to_lds_b128(
        (as1_v4i_p)(grow + e), (as3_v4i_p)(ldsBase + c * kStride + w), 0, 0);
  }
  wait_async0();

  const float4* lp4 = reinterpret_cast<const float4*>(ldsBase + lane * kStride);
  #pragma unroll
  for (int i = 0; i < 8; ++i) {
    float4 q = lp4[i];
    xv[4 * i + 0] = q.x; xv[4 * i + 1] = q.y;
    xv[4 * i + 2] = q.z; xv[4 * i + 3] = q.w;
  }
#else
  const float4* gp4 = reinterpret_cast<const float4*>(grow + lane * kChunk);
  #pragma unroll
  for (int i = 0; i < 8; ++i) {
    float4 q = gp4[i];
    xv[4 * i + 0] = q.x; xv[4 * i + 1] = q.y;
    xv[4 * i + 2] = q.z; xv[4 * i + 3] = q.w;
  }
#endif

  // --- squares (cached) + per-lane chunk sum ---
  // buf first holds x^2; the reverse sweep then overwrites it in place with
  // the output angles (slot k+1 is dead once iteration k+1 has consumed it).
  float buf[kChunk];
  float csum = 0.0f;
  #pragma unroll
  for (int k = 0; k < kChunk; ++k) {
    buf[k] = xv[k] * xv[k];
    csum += buf[k];
  }

  // --- wave inclusive SUFFIX scan of chunk sums (5 shuffle stages) ---
  float inc = csum;
  #pragma unroll
  for (int off = 1; off < 32; off <<= 1) {
    float o = __shfl_down(inc, off, 32);
    inc += (lane + off < 32) ? o : 0.0f;
  }
  float run = __shfl_down(inc, 1, 32);  // exclusive suffix
  if (lane == 31) run = 0.0f;

  // --- fused reverse sweep: rsqrt(suffix sum) + shifted acos ---
  // out[g], g = lane*32 + k + 1, gets acos(x[g-1] * rsqrt(sumsq[g-1:])).
  float a31 = 0.0f;  // k=31 angle -> next lane's slot 0
  float r0  = 0.0f;  // lane 0: full row norm
  const float xLast = xv[kChunk - 1];
  #pragma unroll
  for (int k = kChunk - 1; k >= 0; --k) {
    run += buf[k];
    float rr = dev_rsqf(run);
    rr = rr * fmaf(-0.5f * run * rr, rr, 1.5f);   // 1 Newton step: ~1 ulp
    const float a = fast_acosf(xv[k] * rr);
    if (k == kChunk - 1) a31 = a; else buf[k + 1] = a;
    if (k == 0) r0 = run * rr;                    // sqrt(run)
  }
  const float fromPrev = __shfl_up(a31, 1, 32);
  buf[0] = (lane == 0) ? r0 : fromPrev;           // out[0] = r
  if (lane == 31 && xLast < 0.0f)                 // flip last angle (g=1023)
    buf[kChunk - 1] = kTwoPi - buf[kChunk - 1];

#if USE_ASYNC_LDS
  float4* sp4 = reinterpret_cast<float4*>(ldsBase + lane * kStride);
  #pragma unroll
  for (int i = 0; i < 8; ++i)
    sp4[i] = make_float4(buf[4 * i + 0], buf[4 * i + 1],
                         buf[4 * i + 2], buf[4 * i + 3]);
  wait_ds0();  // LDS writes committed before the async engine reads them
  #pragma unroll
  for (int s = 0; s < 8; ++s) {
    const int e = s * 128 + lane * 4;
    const int c = e >> 5;
    const int w = e & 31;
    __builtin_amdgcn_global_store_async_from_lds_b128(
        (as1_v4i_p)(orow + e), (as3_v4i_p)(ldsBase + c * kStride + w), 0, 0);
  }
  wait_async0();  // drain DMA before LDS retires
#else
  float4* op4 = reinterpret_cast<float4*>(orow + lane * kChunk);
  #pragma unroll
  for (int i = 0; i < 8; ++i)
    op4[i] = make_float4(buf[4 * i + 0], buf[4 * i + 1],
                         buf[4 * i + 2], buf[4 * i + 3]);
#endif
}

extern "C" void kernel_launch(void* const* d_in, const int* in_sizes, int n_in,
                              void* d_out, int out_size, void* d_ws, size_t ws_size,
                              hipStream_t stream) {
  const float* x = (const float*)d_in[0];
  float* out = (float*)d_out;
  const int rows = in_sizes[0] / kD;                // 65536
  const int blocks = (rows + kWaves - 1) / kWaves;  // 8192
  tohypersphere_kernel<<<blocks, 256, 0, stream>>>(x, out, rows);
}